// STAGNN_86045374808292
// MI455X (gfx1250) — compile-verified
//
#include <hip/hip_runtime.h>

#define N_NODES 50000
#define E_EDGES 800000
#define IN_DIM  128
#define HID     64
#define HEADS   4
#define FEAT    (HEADS * HID)   // 256
#define OUT_DIM 40
#define OUT_PAD 48
#define NEG_SLOPE 0.2f

typedef __attribute__((ext_vector_type(2))) float v2f;
typedef __attribute__((ext_vector_type(8))) float v8f;

__device__ __forceinline__ v8f wmma4(v2f a, v2f b, v8f c) {
  // V_WMMA_F32_16X16X4_F32 : D = A(16x4) * B(4x16) + C
  return __builtin_amdgcn_wmma_f32_16x16x4_f32(false, a, false, b, (short)0, c,
                                               false, false);
}

__device__ __forceinline__ float frelu(float v) { return v > 0.f ? v : 0.f; }
__device__ __forceinline__ float lrelu(float v) { return v > 0.f ? v : NEG_SLOPE * v; }

// ---------------------------------------------------------------------------
// GEMM1: h0 = relu(x @ W_proj + b_proj) + time_emb[timesteps]
// A-frags (K=128 -> 32 x v2f) preloaded once; inner loop = ds_load + wmma.
// ---------------------------------------------------------------------------
__global__ void k_gemm1(const float* __restrict__ x, const float* __restrict__ Wp,
                        const float* __restrict__ bp, const float* __restrict__ temb,
                        const int* __restrict__ ts, float* __restrict__ h0) {
  __shared__ float lw[IN_DIM * HID];  // 32 KB
  for (int i = threadIdx.x; i < IN_DIM * HID; i += 256) lw[i] = Wp[i];
  __syncthreads();

  const int lane = threadIdx.x & 31, wave = threadIdx.x >> 5;
  const int lh = lane & 15, kg = lane >> 4;
  const int rowBase = (blockIdx.x * 8 + wave) * 16;
  const int rowA = rowBase + lh;
  // Clamp: rows >= N compute garbage that is never stored -> branch-free loads.
  const int rowC = rowA < N_NODES ? rowA : (N_NODES - 1);

  v2f af[IN_DIM / 4];
  const float* xp = x + rowC * IN_DIM + 2 * kg;
#pragma unroll
  for (int ks = 0; ks < IN_DIM / 4; ++ks) {
    const float2 t = *(const float2*)(xp + ks * 4);
    af[ks][0] = t.x; af[ks][1] = t.y;
  }

  for (int ct = 0; ct < HID / 16; ++ct) {
    const int colB = ct * 16 + lh;
    v8f acc = {0.f, 0.f, 0.f, 0.f, 0.f, 0.f, 0.f, 0.f};
#pragma unroll
    for (int ks = 0; ks < IN_DIM / 4; ++ks) {
      const int kb = ks * 4 + 2 * kg;
      v2f b = {lw[kb * HID + colB], lw[(kb + 1) * HID + colB]};
      acc = wmma4(af[ks], b, acc);
    }
#pragma unroll
    for (int j = 0; j < 8; ++j) {
      const int row = rowBase + j + 8 * kg;
      if (row < N_NODES) {
        float v = frelu(acc[j] + bp[colB]);
        v += temb[ts[row] * HID + colB];
        h0[row * HID + colB] = v;
      }
    }
  }
}

// ---------------------------------------------------------------------------
// GEMM2: h = h0 @ W_gat   (raw).  K=64 -> 16 x v2f A cache, 16 col tiles.
// ---------------------------------------------------------------------------
__global__ void k_gemm2(const float* __restrict__ h0, const float* __restrict__ Wg,
                        float* __restrict__ h) {
  __shared__ float lw[HID * FEAT];  // 64 KB
  for (int i = threadIdx.x; i < HID * FEAT; i += 256) lw[i] = Wg[i];
  __syncthreads();

  const int lane = threadIdx.x & 31, wave = threadIdx.x >> 5;
  const int lh = lane & 15, kg = lane >> 4;
  const int rowBase = (blockIdx.x * 8 + wave) * 16;
  const int rowA = rowBase + lh;
  const int rowC = rowA < N_NODES ? rowA : (N_NODES - 1);

  v2f af[HID / 4];
  const float* xp = h0 + rowC * HID + 2 * kg;
#pragma unroll
  for (int ks = 0; ks < HID / 4; ++ks) {
    const float2 t = *(const float2*)(xp + ks * 4);
    af[ks][0] = t.x; af[ks][1] = t.y;
  }

  for (int ct = 0; ct < FEAT / 16; ++ct) {
    const int colB = ct * 16 + lh;
    v8f acc = {0.f, 0.f, 0.f, 0.f, 0.f, 0.f, 0.f, 0.f};
#pragma unroll
    for (int ks = 0; ks < HID / 4; ++ks) {
      const int kb = ks * 4 + 2 * kg;
      v2f b = {lw[kb * FEAT + colB], lw[(kb + 1) * FEAT + colB]};
      acc = wmma4(af[ks], b, acc);
    }
#pragma unroll
    for (int j = 0; j < 8; ++j) {
      const int row = rowBase + j + 8 * kg;
      if (row < N_NODES) h[row * FEAT + colB] = acc[j];
    }
  }
}

// ---------------------------------------------------------------------------
// Attention coefficients: a_src[n,h] = dot(h[n,h,:], att_src[h,:]) ; same dst
// ---------------------------------------------------------------------------
__global__ void k_att(const float* __restrict__ h, const float* __restrict__ as,
                      const float* __restrict__ ad, float* __restrict__ a_src,
                      float* __restrict__ a_dst) {
  const int lane = threadIdx.x & 31, wave = threadIdx.x >> 5;
  const int node = blockIdx.x * 8 + wave;
  if (node >= N_NODES) return;
  const int head = lane >> 3;
  const float4* hp = (const float4*)(h + node * FEAT + lane * 8);
  const float4* sp = (const float4*)(as + head * HID + (lane & 7) * 8);
  const float4* dp = (const float4*)(ad + head * HID + (lane & 7) * 8);
  float4 x0 = hp[0], x1 = hp[1];
  float4 s0 = sp[0], s1 = sp[1];
  float4 d0 = dp[0], d1 = dp[1];
  float ps = x0.x * s0.x + x0.y * s0.y + x0.z * s0.z + x0.w * s0.w +
             x1.x * s1.x + x1.y * s1.y + x1.z * s1.z + x1.w * s1.w;
  float pd = x0.x * d0.x + x0.y * d0.y + x0.z * d0.z + x0.w * d0.w +
             x1.x * d1.x + x1.y * d1.y + x1.z * d1.z + x1.w * d1.w;
#pragma unroll
  for (int off = 1; off < 8; off <<= 1) {
    ps += __shfl_xor(ps, off, 32);
    pd += __shfl_xor(pd, off, 32);
  }
  if ((lane & 7) == 0) {
    a_src[node * HEADS + head] = ps;
    a_dst[node * HEADS + head] = pd;
  }
}

// ---------------------------------------------------------------------------
// Init: m = -inf, denom = 0, agg = 0
// ---------------------------------------------------------------------------
__global__ void k_init(float* __restrict__ m, float* __restrict__ denom,
                       float* __restrict__ agg) {
  const int idx = blockIdx.x * blockDim.x + threadIdx.x;
  const int stride = gridDim.x * blockDim.x;
  for (int i = idx; i < N_NODES * FEAT; i += stride) agg[i] = 0.f;
  for (int i = idx; i < N_NODES * HEADS; i += stride) {
    m[i] = __int_as_float(0xff800000);  // -inf
    denom[i] = 0.f;
  }
}

__device__ __forceinline__ void load_edge(const int* __restrict__ ei, int e,
                                          int& s, int& d) {
  if (e < E_EDGES) { s = ei[e]; d = ei[E_EDGES + e]; }
  else             { s = d = e - E_EDGES; }  // self loops
}

// ---------------------------------------------------------------------------
// Pass A: segment max (exact float atomic max via signed/unsigned trick)
// ---------------------------------------------------------------------------
__global__ void k_edge_max(const int* __restrict__ ei, const float* __restrict__ a_src,
                           const float* __restrict__ a_dst, float* __restrict__ m) {
  const int e = blockIdx.x * blockDim.x + threadIdx.x;
  if (e >= E_EDGES + N_NODES) return;
  int s, d; load_edge(ei, e, s, d);
#pragma unroll
  for (int h = 0; h < HEADS; ++h) {
    const float v = lrelu(a_src[s * HEADS + h] + a_dst[d * HEADS + h]);
    float* addr = m + d * HEADS + h;
    if (v >= 0.f) atomicMax((int*)addr, __float_as_int(v));
    else          atomicMin((unsigned int*)addr, (unsigned int)__float_as_int(v));
  }
}

// ---------------------------------------------------------------------------
// Pass B: segment sum of exp(e - max)
// ---------------------------------------------------------------------------
__global__ void k_edge_sum(const int* __restrict__ ei, const float* __restrict__ a_src,
                           const float* __restrict__ a_dst, const float* __restrict__ m,
                           float* __restrict__ denom) {
  const int e = blockIdx.x * blockDim.x + threadIdx.x;
  if (e >= E_EDGES + N_NODES) return;
  int s, d; load_edge(ei, e, s, d);
#pragma unroll
  for (int h = 0; h < HEADS; ++h) {
    const float v = lrelu(a_src[s * HEADS + h] + a_dst[d * HEADS + h]);
    atomicAdd(denom + d * HEADS + h, expf(v - m[d * HEADS + h]));
  }
}

// ---------------------------------------------------------------------------
// Pass C: agg[dst] += alpha * h[src]. One wave per edge; 8 coalesced
// f32 atomics per lane (adjacent lanes -> adjacent addresses, L2-resident).
// ---------------------------------------------------------------------------
__global__ void k_edge_agg(const int* __restrict__ ei, const float* __restrict__ a_src,
                           const float* __restrict__ a_dst, const float* __restrict__ m,
                           const float* __restrict__ denom, const float* __restrict__ h,
                           float* __restrict__ agg) {
  const int lane = threadIdx.x & 31, wave = threadIdx.x >> 5;
  const int e = blockIdx.x * 8 + wave;
  if (e >= E_EDGES + N_NODES) return;
  int s, d; load_edge(ei, e, s, d);
  float av = 0.f;
  if (lane < HEADS) {
    const float v = lrelu(a_src[s * HEADS + lane] + a_dst[d * HEADS + lane]);
    av = expf(v - m[d * HEADS + lane]) / (denom[d * HEADS + lane] + 1e-16f);
  }
  float alpha[HEADS];
#pragma unroll
  for (int hh = 0; hh < HEADS; ++hh) alpha[hh] = __shfl(av, hh, 32);
  const float* hs = h + s * FEAT;
  float* ap = agg + d * FEAT;
#pragma unroll
  for (int i = 0; i < FEAT / 32; ++i) {
    const int col = lane + 32 * i;
    atomicAdd(ap + col, alpha[col >> 6] * hs[col]);
  }
}

// ---------------------------------------------------------------------------
// GEMM3: out = relu(agg + b_gat) @ W_cls + b_cls   (W padded 40 -> 48 cols)
// K=256 streamed in 4 chunks of 16 k-steps; 3 accumulators held live.
// ---------------------------------------------------------------------------
__global__ void k_gemm3(const float* __restrict__ agg, const float* __restrict__ bg,
                        const float* __restrict__ Wc, const float* __restrict__ bc,
                        float* __restrict__ out) {
  __shared__ float lw[FEAT * OUT_PAD];  // 48 KB
  __shared__ float lbg[FEAT];           // 1 KB
  for (int i = threadIdx.x; i < FEAT * OUT_PAD; i += 256) {
    const int k = i / OUT_PAD, c = i % OUT_PAD;
    lw[i] = (c < OUT_DIM) ? Wc[k * OUT_DIM + c] : 0.f;
  }
  for (int i = threadIdx.x; i < FEAT; i += 256) lbg[i] = bg[i];
  __syncthreads();

  const int lane = threadIdx.x & 31, wave = threadIdx.x >> 5;
  const int lh = lane & 15, kg = lane >> 4;
  const int rowBase = (blockIdx.x * 8 + wave) * 16;
  const int rowA = rowBase + lh;
  const int rowC = rowA < N_NODES ? rowA : (N_NODES - 1);
  const float* ap = agg + rowC * FEAT;

  v8f acc[3];
#pragma unroll
  for (int ct = 0; ct < 3; ++ct) acc[ct] = (v8f){0.f, 0.f, 0.f, 0.f, 0.f, 0.f, 0.f, 0.f};

  for (int chunk = 0; chunk < FEAT / 64; ++chunk) {
    v2f af[16];
#pragma unroll
    for (int ks = 0; ks < 16; ++ks) {
      const int kb = chunk * 64 + ks * 4 + 2 * kg;
      const float2 t = *(const float2*)(ap + kb);
      af[ks][0] = frelu(t.x + lbg[kb]);
      af[ks][1] = frelu(t.y + lbg[kb + 1]);
    }
#pragma unroll
    for (int ct = 0; ct < 3; ++ct) {
      const int colB = ct * 16 + lh;
#pragma unroll
      for (int ks = 0; ks < 16; ++ks) {
        const int kb = chunk * 64 + ks * 4 + 2 * kg;
        v2f b = {lw[kb * OUT_PAD + colB], lw[(kb + 1) * OUT_PAD + colB]};
        acc[ct] = wmma4(af[ks], b, acc[ct]);
      }
    }
  }

#pragma unroll
  for (int ct = 0; ct < 3; ++ct) {
    const int colB = ct * 16 + lh;
    if (colB < OUT_DIM) {
#pragma unroll
      for (int j = 0; j < 8; ++j) {
        const int row = rowBase + j + 8 * kg;
        if (row < N_NODES) out[row * OUT_DIM + colB] = acc[ct][j] + bc[colB];
      }
    }
  }
}

// ---------------------------------------------------------------------------
extern "C" void kernel_launch(void* const* d_in, const int* in_sizes, int n_in,
                              void* d_out, int out_size, void* d_ws, size_t ws_size,
                              hipStream_t stream) {
  (void)in_sizes; (void)n_in; (void)out_size; (void)ws_size;
  const float* x    = (const float*)d_in[0];
  const int*   ei   = (const int*)d_in[1];
  const int*   ts   = (const int*)d_in[2];
  const float* Wp   = (const float*)d_in[3];
  const float* bp   = (const float*)d_in[4];
  const float* temb = (const float*)d_in[5];
  const float* Wg   = (const float*)d_in[6];
  const float* as   = (const float*)d_in[7];
  const float* ad   = (const float*)d_in[8];
  const float* bg   = (const float*)d_in[9];
  const float* Wc   = (const float*)d_in[10];
  const float* bc   = (const float*)d_in[11];
  float* out = (float*)d_out;

  float* ws = (float*)d_ws;
  float* h0    = ws; ws += N_NODES * HID;    // 12.8 MB
  float* hfeat = ws; ws += N_NODES * FEAT;   // 51.2 MB (L2-resident)
  float* a_src = ws; ws += N_NODES * HEADS;
  float* a_dst = ws; ws += N_NODES * HEADS;
  float* m     = ws; ws += N_NODES * HEADS;
  float* denom = ws; ws += N_NODES * HEADS;
  float* agg   = ws; ws += N_NODES * FEAT;   // 51.2 MB (L2-resident)

  const int gemmGrid = (N_NODES + 127) / 128;  // 8 waves x 16 rows per block
  const int EN = E_EDGES + N_NODES;

  k_gemm1<<<gemmGrid, 256, 0, stream>>>(x, Wp, bp, temb, ts, h0);
  k_gemm2<<<gemmGrid, 256, 0, stream>>>(h0, Wg, hfeat);
  k_att<<<(N_NODES + 7) / 8, 256, 0, stream>>>(hfeat, as, ad, a_src, a_dst);
  k_init<<<1024, 256, 0, stream>>>(m, denom, agg);
  k_edge_max<<<(EN + 255) / 256, 256, 0, stream>>>(ei, a_src, a_dst, m);
  k_edge_sum<<<(EN + 255) / 256, 256, 0, stream>>>(ei, a_src, a_dst, m, denom);
  k_edge_agg<<<(EN + 7) / 8, 256, 0, stream>>>(ei, a_src, a_dst, m, denom, hfeat, agg);
  k_gemm3<<<gemmGrid, 256, 0, stream>>>(agg, bg, Wc, bc, out);
}